// RandomTimeStretch_50861002719980
// MI455X (gfx1250) — compile-verified
//
#include <hip/hip_runtime.h>
#include <hip/hip_bf16.h>
#include <math.h>

// Phase vocoder + abs() == linear magnitude interpolation (phase cancels).
// Bandwidth-bound: ~380 MB traffic -> ~16 us floor at 23.3 TB/s on MI455X.
// Strategy: one block per (batch, freq) row; stage the 2048-sample complex row
// into LDS via CDNA5 async global->LDS B128 DMA, then gather/interp from LDS.

#define TPB     256
#define T_IN    2048
#define RATE_F  1.2f
#define ROW_PAD (T_IN + 2)   // torchaudio pads time dim by 2 zeros

#if __has_builtin(__builtin_amdgcn_global_load_async_to_lds_b128)
#define USE_ASYNC_LDS 1
#else
#define USE_ASYNC_LDS 0
#endif

// Builtin signature (from clang diagnostic): param 0 is
// 'int __vector(4) __device__ *'  == v4i in addrspace(1), non-const.
typedef int v4i __attribute__((vector_size(16)));
typedef __attribute__((address_space(1))) v4i* GlobalV4P;
typedef __attribute__((address_space(3))) v4i* LdsV4P;

__global__ __launch_bounds__(TPB)
void pv_mag_kernel(const float* __restrict__ re,
                   const float* __restrict__ im,
                   float* __restrict__ out,
                   int t_out)
{
    __shared__ __align__(16) float s_re[ROW_PAD];
    __shared__ __align__(16) float s_im[ROW_PAD];

    const int row = blockIdx.x;          // row = b * N_FREQ + f
    const int tid = threadIdx.x;
    const float* re_row = re + (size_t)row * T_IN;
    const float* im_row = im + (size_t)row * T_IN;

    // ---- Stage the whole row (re + im) into LDS, 16B per lane per op ----
    const int nvec = T_IN / 4;           // 512 float4 chunks per array
#if USE_ASYNC_LDS
    for (int i = tid; i < nvec; i += TPB) {
        __builtin_amdgcn_global_load_async_to_lds_b128(
            (GlobalV4P)(re_row + 4 * i), (LdsV4P)(s_re + 4 * i), 0, 0);
        __builtin_amdgcn_global_load_async_to_lds_b128(
            (GlobalV4P)(im_row + 4 * i), (LdsV4P)(s_im + 4 * i), 0, 0);
    }
#else
    {
        const float4* g_re = (const float4*)re_row;
        const float4* g_im = (const float4*)im_row;
        float4* l_re = (float4*)s_re;
        float4* l_im = (float4*)s_im;
        for (int i = tid; i < nvec; i += TPB) {
            l_re[i] = g_re[i];
            l_im[i] = g_im[i];
        }
    }
#endif
    if (tid < 2) {                       // zero the 2-sample pad tail
        s_re[T_IN + tid] = 0.0f;
        s_im[T_IN + tid] = 0.0f;
    }
#if USE_ASYNC_LDS
#if __has_builtin(__builtin_amdgcn_s_wait_asynccnt)
    __builtin_amdgcn_s_wait_asynccnt(0);
#else
    asm volatile("s_wait_asynccnt 0" ::: "memory");
#endif
#endif
    __syncthreads();

    // ---- Gather + magnitude lerp; coalesced stores ----
    float* out_row = out + (size_t)row * t_out;
    for (int t = tid; t < t_out; t += TPB) {
        float ts  = (float)t * RATE_F;   // matches arange(0, T, 1.2)[t]
        int   idx = (int)ts;             // floor (ts >= 0)
        float a   = ts - (float)idx;     // alpha = ts % 1.0
        float re0 = s_re[idx],     im0 = s_im[idx];
        float re1 = s_re[idx + 1], im1 = s_im[idx + 1];
        float n0 = sqrtf(fmaf(re0, re0, im0 * im0));
        float n1 = sqrtf(fmaf(re1, re1, im1 * im1));
        // alpha*n1 + (1-alpha)*n0
        out_row[t] = fmaf(a, n1 - n0, n0);
    }
}

extern "C" void kernel_launch(void* const* d_in, const int* in_sizes, int n_in,
                              void* d_out, int out_size, void* d_ws, size_t ws_size,
                              hipStream_t stream) {
    const float* re = (const float*)d_in[0];
    const float* im = (const float*)d_in[1];
    float* out = (float*)d_out;

    const int rows  = in_sizes[0] / T_IN;   // B * N_FREQ = 16 * 1025 = 16400
    const int t_out = out_size / rows;      // 1707

    pv_mag_kernel<<<rows, TPB, 0, stream>>>(re, im, out, t_out);
}